// SampleAndGroup_16776142258204
// MI455X (gfx1250) — compile-verified
//
#include <hip/hip_runtime.h>
#include <hip/hip_bf16.h>

// ---- problem constants (from reference) ----
#define NB      8
#define NPTS    4096
#define NPOINT  1024
#define NSAMPLE 32
#define NCH     64
#define R2      0.16f          // RADIUS^2 (compare in squared-distance space)
#define OUTCH   (3 + NCH)      // 67
#define NTILE   ((NPTS / 16) * (NPTS / 16))   // 65536 16x16 tiles
#define TPW     (NTILE / 32)                  // 2048 tiles per wave

typedef float v2f __attribute__((ext_vector_type(2)));
typedef float v8f __attribute__((ext_vector_type(8)));
typedef unsigned long long u64;

// pack (non-negative float value, index) into a u64 key:
//   max(key)  ==  max value, ties broken toward SMALLEST index (argmax semantics)
__device__ __forceinline__ u64 pack_key(float v, unsigned idx) {
    return ((u64)__float_as_uint(v) << 32) | (u64)(~idx);
}

// =====================================================================
// Kernel 1: per-batch fused seed-search (WMMA) + farthest point sampling
//   one workgroup (1024 threads = 32 wave32) per batch
//   LDS: coords 48KB + fps list 4KB + key scratch (< 64KB total)
// =====================================================================
__global__ __launch_bounds__(1024) void fps_kernel(
    const float* __restrict__ xyz,     // (B, N, 3)
    float* __restrict__ out_newxyz,    // (B, NPOINT, 3)
    int* __restrict__ fps_idx)         // (B, NPOINT) in workspace
{
    __shared__ float xs[NPTS], ys[NPTS], zs[NPTS];
    __shared__ u64   wkey[32];
    __shared__ int   fpsLds[NPOINT];
    __shared__ float selx, sely, selz;

    const int b    = blockIdx.x;
    const int tid  = threadIdx.x;
    const int lane = tid & 31;
    const int wave = tid >> 5;
    const float* X = xyz + (size_t)b * NPTS * 3;

    // stage coords into LDS (coalesced)
    for (int j = tid; j < NPTS; j += 1024) {
        xs[j] = X[j * 3 + 0];
        ys[j] = X[j * 3 + 1];
        zs[j] = X[j * 3 + 2];
    }
    __syncthreads();

    // ---------------- seed: argmax over full d2 matrix via WMMA ----------------
    // D = X * X^T as 16x16 tiles of V_WMMA_F32_16X16X4_F32 (K=3 padded to 4);
    // d2(m,n) = |xm|^2 + |xn|^2 - 2*D[m][n]; branchless packed-key argmax.
    u64 best = 0;
    const int  r16 = lane & 15;
    const bool lo  = lane < 16;

#pragma unroll 2
    for (int t = wave * TPW; t < (wave + 1) * TPW; ++t) {
        const int ti = t >> 8, tj = t & 255;
        const int m0 = ti * 16, n0 = tj * 16;
        const int ma = m0 + r16;        // A-row owned by this lane
        const int nb = n0 + r16;        // B-col owned by this lane

        // A 16x4 f32 layout: lanes 0-15 hold K0,K1; lanes 16-31 hold K2,K3(=0 pad)
        v2f a, bm;
        a.x  = lo ? xs[ma] : zs[ma];
        a.y  = lo ? ys[ma] : 0.0f;
        bm.x = lo ? xs[nb] : zs[nb];
        bm.y = lo ? ys[nb] : 0.0f;

        v8f c = {};
        c = __builtin_amdgcn_wmma_f32_16x16x4_f32(
                false, a, false, bm, (short)0, c, false, false);

        // C layout: vgpr r, lanes 0-15 -> (M=r, N=lane); lanes 16-31 -> (M=r+8, N=lane-16)
        const int   n  = n0 + r16;
        const float nn = xs[n] * xs[n] + ys[n] * ys[n] + zs[n] * zs[n];
#pragma unroll
        for (int r = 0; r < 8; ++r) {
            const int   m  = m0 + r + (lo ? 0 : 8);
            const float nm = xs[m] * xs[m] + ys[m] * ys[m] + zs[m] * zs[m];
            const float d2 = fmaxf(nm + nn - 2.0f * c[r], 0.0f); // clamp: keep sign bit clear
            const u64   k  = pack_key(d2, (unsigned)(m * NPTS + n));
            best = k > best ? k : best;                           // branchless
        }
    }
    // wave32 reduction (branchless 64-bit max)
    for (int off = 16; off; off >>= 1) {
        const u64 ok = __shfl_xor(best, off, 32);
        best = ok > best ? ok : best;
    }
    if (lane == 0) wkey[wave] = best;
    __syncthreads();
    if (tid == 0) {
        u64 k = wkey[0];
        for (int w = 1; w < 32; ++w) k = wkey[w] > k ? wkey[w] : k;
        const unsigned fl = ~(unsigned)(k & 0xffffffffu);
        const int j0 = (int)(fl / NPTS);   // row  (== flat // N in reference)
        const int i0 = (int)(fl % NPTS);   // col
        fpsLds[0] = j0; fpsLds[1] = i0;
        fps_idx[b * NPOINT + 0] = j0;
        fps_idx[b * NPOINT + 1] = i0;
    }
    __syncthreads();

    // ---------------- FPS: dist init + 1022 serial steps ----------------
    const int j0 = fpsLds[0], i0 = fpsLds[1];
    const float xi0 = xs[i0], yi0 = ys[i0], zi0 = zs[i0];
    const float xj0 = xs[j0], yj0 = ys[j0], zj0 = zs[j0];

    float dist[4];
#pragma unroll
    for (int k = 0; k < 4; ++k) {
        const int j = tid * 4 + k;
        const float dxa = xs[j] - xi0, dya = ys[j] - yi0, dza = zs[j] - zi0;
        const float dxb = xs[j] - xj0, dyb = ys[j] - yj0, dzb = zs[j] - zj0;
        dist[k] = fminf(dxa * dxa + dya * dya + dza * dza,
                        dxb * dxb + dyb * dyb + dzb * dzb);
    }

    for (int s = 2; s < NPOINT; ++s) {
        // branchless local + wave argmax via packed keys
        u64 k0 = pack_key(dist[0], (unsigned)(tid * 4 + 0));
#pragma unroll
        for (int k = 1; k < 4; ++k) {
            const u64 kk = pack_key(dist[k], (unsigned)(tid * 4 + k));
            k0 = kk > k0 ? kk : k0;
        }
        for (int off = 16; off; off >>= 1) {
            const u64 ok = __shfl_xor(k0, off, 32);
            k0 = ok > k0 ? ok : k0;
        }
        if (lane == 0) wkey[wave] = k0;
        __syncthreads();
        if (tid == 0) {
            u64 kk = wkey[0];
            for (int w = 1; w < 32; ++w) kk = wkey[w] > kk ? wkey[w] : kk;
            const int mi = (int)(~(unsigned)(kk & 0xffffffffu));
            fpsLds[s] = mi;
            fps_idx[b * NPOINT + s] = mi;
            selx = xs[mi]; sely = ys[mi]; selz = zs[mi];
        }
        __syncthreads();
        const float sx = selx, sy = sely, sz = selz;
#pragma unroll
        for (int k = 0; k < 4; ++k) {
            const int j = tid * 4 + k;
            const float dx = xs[j] - sx, dy = ys[j] - sy, dz = zs[j] - sz;
            dist[k] = fminf(dist[k], dx * dx + dy * dy + dz * dz);
        }
    }

    // new_xyz = xyz gathered at fps indices
    if (tid < NPOINT) {
        const int idx = fpsLds[tid];
        float* o = out_newxyz + ((size_t)b * NPOINT + tid) * 3;
        o[0] = xs[idx]; o[1] = ys[idx]; o[2] = zs[idx];
    }
}

// =====================================================================
// Kernel 2: ball query + grouping — one wave32 per center (NSAMPLE==32)
// =====================================================================
__global__ __launch_bounds__(256) void ballgroup_kernel(
    const float* __restrict__ xyz,     // (B, N, 3)
    const float* __restrict__ pts,     // (B, N, 64)
    const int* __restrict__ fps_idx,   // (B, NPOINT)
    float* __restrict__ out_np)        // (B, NPOINT, NSAMPLE, 67)
{
    __shared__ int idxbuf[8][NSAMPLE];

    const int lane = threadIdx.x & 31;
    const int w    = threadIdx.x >> 5;
    const int gw   = blockIdx.x * 8 + w;       // global wave id = center id
    const int b    = gw >> 10;
    const int c    = gw & (NPOINT - 1);

    const int ci = fps_idx[b * NPOINT + c];
    const float* X = xyz + (size_t)b * NPTS * 3;
    const float cx = X[ci * 3 + 0], cy = X[ci * 3 + 1], cz = X[ci * 3 + 2];

    // ball query: first NSAMPLE indices (ascending) with d2 < R2; pad with first hit
    unsigned cnt = 0;
    int first = 0;
    bool haveFirst = false;
    for (int base = 0; base < NPTS && cnt < NSAMPLE; base += 32) {
        const int j = base + lane;
        const float dx = X[j * 3 + 0] - cx;
        const float dy = X[j * 3 + 1] - cy;
        const float dz = X[j * 3 + 2] - cz;
        const bool inball = (dx * dx + dy * dy + dz * dz) < R2;
        const unsigned m = (unsigned)__ballot(inball);
        if (!haveFirst && m) { first = base + (__ffs((int)m) - 1); haveFirst = true; }
        const unsigned pos = cnt + (unsigned)__popc(m & ((1u << lane) - 1u));
        if (inball && pos < NSAMPLE) idxbuf[w][pos] = j;
        cnt += (unsigned)__popc(m);
    }
    const unsigned c32 = cnt < NSAMPLE ? cnt : NSAMPLE;
    if ((unsigned)lane >= c32) idxbuf[w][lane] = first;   // pad (0 if no hits)
    __syncthreads();

    // grouping: channels 0..2 = xyz - center, 3..66 = point features
    const float* P   = pts + (size_t)b * NPTS * NCH;
    float* dst0 = out_np + ((size_t)(b * NPOINT + c) * NSAMPLE) * OUTCH;
    const float cc = (lane == 0) ? cx : (lane == 1) ? cy : cz;

    for (int s = 0; s < NSAMPLE; ++s) {
        const int idx = idxbuf[w][s];
        float* dst = dst0 + s * OUTCH;
        if (lane < 3) dst[lane] = X[idx * 3 + lane] - cc;
        dst[3 + lane]      = P[idx * NCH + lane];
        dst[3 + 32 + lane] = P[idx * NCH + 32 + lane];
    }
}

// =====================================================================
extern "C" void kernel_launch(void* const* d_in, const int* in_sizes, int n_in,
                              void* d_out, int out_size, void* d_ws, size_t ws_size,
                              hipStream_t stream)
{
    (void)in_sizes; (void)n_in; (void)out_size; (void)ws_size;

    const float* xyz = (const float*)d_in[0];   // (8, 4096, 3)
    const float* pts = (const float*)d_in[1];   // (8, 4096, 64)

    float* out        = (float*)d_out;
    float* out_newxyz = out;                                   // (8,1024,3)
    float* out_np     = out + (size_t)NB * NPOINT * 3;         // (8,1024,32,67)
    int*   fps        = (int*)d_ws;                            // (8,1024) indices

    fps_kernel<<<NB, 1024, 0, stream>>>(xyz, out_newxyz, fps);
    ballgroup_kernel<<<(NB * NPOINT) / 8, 256, 0, stream>>>(xyz, pts, fps, out_np);
}